// SridharGCN1dBlock_11751030522224
// MI455X (gfx1250) — compile-verified
//
#include <hip/hip_runtime.h>

// Problem constants (match reference)
#define GNUM 64     // B*N graphs
#define BNUM 4
#define NSET 16
#define C0IN 64
#define CC   128
#define LL   2048
#define EE   16384
#define EPSF 1e-5f

typedef __attribute__((ext_vector_type(2))) float v2f;
typedef __attribute__((ext_vector_type(8))) float v8f;

#if defined(__HIP_DEVICE_COMPILE__) && __has_builtin(__builtin_amdgcn_global_load_async_to_lds_b128)
#define USE_ASYNC_LDS 1
typedef int v4i_vs __attribute__((vector_size(16)));
typedef __attribute__((address_space(1))) v4i_vs* as1_v4i;
typedef __attribute__((address_space(3))) v4i_vs* as3_v4i;
#else
#define USE_ASYNC_LDS 0
#endif

__device__ __forceinline__ void atomAddF(float* p, float v) {
  __hip_atomic_fetch_add(p, v, __ATOMIC_RELAXED, __HIP_MEMORY_SCOPE_AGENT);
}

// ---------------- small setup kernels ----------------
__global__ __launch_bounds__(256) void k_zero(float* p, int n) {
  int i = blockIdx.x * 256 + threadIdx.x;
  if (i < n) p[i] = 0.f;
}

__global__ __launch_bounds__(256) void k_deg(const int* __restrict__ dst,
                                             float* __restrict__ deg) {
  int e = blockIdx.x * 256 + threadIdx.x;
  if (e < EE) atomAddF(&deg[dst[e]], 1.0f);
}

__global__ __launch_bounds__(256) void k_dinv(const float* __restrict__ deg,
                                              float* __restrict__ dinv,
                                              float* __restrict__ selfc) {
  int l = blockIdx.x * 256 + threadIdx.x;
  if (l < LL) {
    float di = rsqrtf(deg[l] + 2.0f);   // improved GCN: +2 self loop
    dinv[l] = di;
    selfc[l] = 2.0f * di * di;
  }
}

__global__ __launch_bounds__(256) void k_edgenorm(const int* __restrict__ src,
                                                  const int* __restrict__ dst,
                                                  const float* __restrict__ dinv,
                                                  float* __restrict__ norm) {
  int e = blockIdx.x * 256 + threadIdx.x;
  if (e < EE) norm[e] = dinv[src[e]] * dinv[dst[e]];
}

// ---------------- tiled transpose: per-g [Rr,Cc] -> [Cc,Rr] ----------------
__global__ __launch_bounds__(256) void k_transpose(const float* __restrict__ in,
                                                   float* __restrict__ out,
                                                   int Rr, int Cc) {
  __shared__ float t[32][33];
  const int g = blockIdx.z;
  const int c0 = blockIdx.x * 32;
  const int r0 = blockIdx.y * 32;
  const float* ip = in + (size_t)g * Rr * Cc;
  float* op = out + (size_t)g * Rr * Cc;
  const int tx = threadIdx.x, ty = threadIdx.y;  // 32 x 8
#pragma unroll
  for (int i = 0; i < 32; i += 8)
    t[ty + i][tx] = ip[(size_t)(r0 + ty + i) * Cc + c0 + tx];
  __syncthreads();
#pragma unroll
  for (int i = 0; i < 32; i += 8)
    op[(size_t)(c0 + ty + i) * Rr + r0 + tx] = t[tx][ty + i];
}

// ---------------- fp32 WMMA GEMM + fused self/bias epilogue ----------------
// hw[M,128] = A[M,K] @ W[K,128]      (M = G*L)
// agg[M,128] = selfc[l]*hw + bias    (l = row % L)
// One block = 8 waves covering N=128; block M-tile = 32 rows; each wave runs
// TWO independent accumulator chains (rows 0-15 / 16-31) sharing each B
// fragment -> half the W loads per FLOP and two interleaved XDL chains.
// A tile staged through LDS (padded rows -> conflict-free ds reads) via
// GLOBAL_LOAD_ASYNC_TO_LDS_B128.
template <int K>
__global__ __launch_bounds__(256) void k_gemm_wmma(const float* __restrict__ A,
                                                   const float* __restrict__ W,
                                                   const float* __restrict__ selfc,
                                                   const float* __restrict__ bias,
                                                   float* __restrict__ hw,
                                                   float* __restrict__ agg) {
  constexpr int KP = K + 4;          // padded row stride (banks 0,4,..,60)
  __shared__ float at[32 * KP];
  const int tid = threadIdx.x;
  const size_t m0 = (size_t)blockIdx.x * 32;
  const float* srcA = A + m0 * K;    // A tile [32][K] contiguous in memory

#if USE_ASYNC_LDS
  {
#pragma unroll
    for (int i = tid; i < 32 * K / 4; i += 256) {   // 16-byte chunks
      int r = (i * 4) / K;
      int col = i * 4 - r * K;
      __builtin_amdgcn_global_load_async_to_lds_b128(
          (as1_v4i)(srcA + (size_t)i * 4),
          (as3_v4i)(at + r * KP + col), 0, 0);
    }
#if __has_builtin(__builtin_amdgcn_s_wait_asynccnt)
    __builtin_amdgcn_s_wait_asynccnt(0);
#else
    asm volatile("s_wait_asynccnt 0x0" ::: "memory");
#endif
  }
#else
#pragma unroll
  for (int i = tid; i < 32 * K / 4; i += 256) {
    int r = (i * 4) / K;
    int col = i * 4 - r * K;
    *(float4*)(at + r * KP + col) = ((const float4*)srcA)[i];
  }
#endif
  __syncthreads();

  const int wave = tid >> 5;       // wave32
  const int lane = tid & 31;
  const int n0 = wave * 16;        // 8 waves cover N=128
  const int mn = lane & 15;        // row for A frag, col for B/D frags
  const int kh = lane >> 4;        // K-half select (lanes 16-31 -> K+2)

  v8f acc0 = {0.f, 0.f, 0.f, 0.f, 0.f, 0.f, 0.f, 0.f};
  v8f acc1 = {0.f, 0.f, 0.f, 0.f, 0.f, 0.f, 0.f, 0.f};
#pragma unroll
  for (int k = 0; k < K; k += 4) {
    v2f a0, a1, b;
    // A 16x4 f32 layout: lanes 0-15: V0=K, V1=K+1 ; lanes 16-31: V0=K+2, V1=K+3
    a0.x = at[mn * KP + k + 2 * kh + 0];
    a0.y = at[mn * KP + k + 2 * kh + 1];
    a1.x = at[(mn + 16) * KP + k + 2 * kh + 0];
    a1.y = at[(mn + 16) * KP + k + 2 * kh + 1];
    // B 4x16 f32 (rows striped across lanes), shared by both chains
    b.x = W[(size_t)(k + 2 * kh + 0) * CC + n0 + mn];
    b.y = W[(size_t)(k + 2 * kh + 1) * CC + n0 + mn];
    acc0 = __builtin_amdgcn_wmma_f32_16x16x4_f32(false, a0, false, b,
                                                 (short)0, acc0, false, false);
    acc1 = __builtin_amdgcn_wmma_f32_16x16x4_f32(false, a1, false, b,
                                                 (short)0, acc1, false, false);
  }
  // C/D 16x16 f32: VGPR r, lane l -> m = r + 8*(l>>4), n = l&15
  float* hp = hw + m0 * CC + n0;
  float* ap = agg + m0 * CC + n0;
  const float bn = bias[n0 + mn];
#pragma unroll
  for (int r = 0; r < 8; ++r) {
    const int mrow = r + 8 * kh;
    {
      const float sc = selfc[((int)m0 + mrow) & (LL - 1)];
      const float v = acc0[r];
      hp[(size_t)mrow * CC + mn] = v;
      ap[(size_t)mrow * CC + mn] = fmaf(sc, v, bn);
    }
    {
      const int mrow1 = mrow + 16;
      const float sc = selfc[((int)m0 + mrow1) & (LL - 1)];
      const float v = acc1[r];
      hp[(size_t)mrow1 * CC + mn] = v;
      ap[(size_t)mrow1 * CC + mn] = fmaf(sc, v, bn);
    }
  }
}

// ---------------- edge scatter: agg[g,dst] += norm[e]*hw[g,src] -----------
// one wave per (g,e); 32 lanes x float4 = 128 channels; stays in L2
__global__ __launch_bounds__(256) void k_scatter(const float4* __restrict__ hw,
                                                 const int* __restrict__ src,
                                                 const int* __restrict__ dst,
                                                 const float* __restrict__ norm,
                                                 float* __restrict__ agg) {
  size_t wid = ((size_t)blockIdx.x * 256 + threadIdx.x) >> 5;
  int lane = threadIdx.x & 31;
  int e = (int)(wid & (EE - 1));
  int g = (int)(wid >> 14);     // EE = 2^14
  int s = src[e], d = dst[e];
  float nm = norm[e];
  float4 v = hw[((size_t)g * LL + s) * 32 + lane];
  float* pd = agg + ((size_t)g * LL + d) * CC + lane * 4;
  atomAddF(pd + 0, v.x * nm);
  atomAddF(pd + 1, v.y * nm);
  atomAddF(pd + 2, v.z * nm);
  atomAddF(pd + 3, v.w * nm);
}

// ------- set-mean over N=16 into mbuf + BN sum/sumsq (no z writeback) -----
__global__ __launch_bounds__(128) void k_center_stats(const float* __restrict__ z,
                                                      float* __restrict__ mbuf,
                                                      float* __restrict__ stats) {
  int c = threadIdx.x;   // channel 0..127
  int l = blockIdx.x;    // 0..2047
  int b = blockIdx.y;    // 0..3
  float v[NSET];
  float m = 0.f;
#pragma unroll
  for (int n = 0; n < NSET; ++n) {
    v[n] = z[(((size_t)(b * NSET + n)) * LL + l) * CC + c];
    m += v[n];
  }
  m *= (1.0f / NSET);
  mbuf[((size_t)b * LL + l) * CC + c] = m;
  float s = 0.f, sq = 0.f;
#pragma unroll
  for (int n = 0; n < NSET; ++n) {
    float zc = v[n] - m;
    s += zc;
    sq += zc * zc;
  }
  atomAddF(&stats[c], s);
  atomAddF(&stats[CC + c], sq);
}

__global__ __launch_bounds__(128) void k_stats_final(float* __restrict__ stats,
                                                     const float* __restrict__ gam,
                                                     const float* __restrict__ bet) {
  int c = threadIdx.x;
  const float inv = 1.0f / (float)(GNUM * LL);
  float mu = stats[c] * inv;
  float var = stats[CC + c] * inv - mu * mu;
  float sc = gam[c] * rsqrtf(var + EPSF);
  stats[2 * CC + c] = sc;
  stats[3 * CC + c] = bet[c] - mu * sc;
}

// ---------------- (v - mean)*scale + shift, ReLU ----------------
__global__ __launch_bounds__(256) void k_apply(const float4* __restrict__ z,
                                               const float4* __restrict__ mbuf,
                                               const float* __restrict__ stats,
                                               float4* __restrict__ out) {
  size_t i = (size_t)blockIdx.x * 256 + threadIdx.x;  // over G*L*32 float4s
  int c4 = (int)(i & 31);
  size_t gl = i >> 5;                  // g*L + l
  int l = (int)(gl & (LL - 1));
  int b = (int)(gl >> 15);             // g = gl>>11, b = g>>4
  float4 m = mbuf[((size_t)b * LL + l) * 32 + c4];
  float4 sc = ((const float4*)(stats + 2 * CC))[c4];
  float4 sh = ((const float4*)(stats + 3 * CC))[c4];
  float4 zz = z[i];
  float4 r;
  r.x = fmaxf(fmaf(zz.x - m.x, sc.x, sh.x), 0.f);
  r.y = fmaxf(fmaf(zz.y - m.y, sc.y, sh.y), 0.f);
  r.z = fmaxf(fmaf(zz.z - m.z, sc.z, sh.z), 0.f);
  r.w = fmaxf(fmaf(zz.w - m.w, sc.w, sh.w), 0.f);
  out[i] = r;
}

// ---------------- orchestration ----------------
extern "C" void kernel_launch(void* const* d_in, const int* in_sizes, int n_in,
                              void* d_out, int out_size, void* d_ws, size_t ws_size,
                              hipStream_t stream) {
  const float* x = (const float*)d_in[0];
  const int* ei = (const int*)d_in[1];    // [2, E], row0=src, row1=dst (int32)
  const int* srcI = ei;
  const int* dstI = ei + EE;
  const float* Wl[3]  = {(const float*)d_in[2],  (const float*)d_in[6],  (const float*)d_in[10]};
  const float* bl[3]  = {(const float*)d_in[3],  (const float*)d_in[7],  (const float*)d_in[11]};
  const float* gml[3] = {(const float*)d_in[4],  (const float*)d_in[8],  (const float*)d_in[12]};
  const float* bel[3] = {(const float*)d_in[5],  (const float*)d_in[9],  (const float*)d_in[13]};

  const size_t big = (size_t)GNUM * LL * CC;  // 16.7M floats = 64 MiB
  float* bufA = (float*)d_ws;        // layer input  [G,L,K]
  float* bufB = bufA + big;          // hw           [G,L,128]
  float* bufC = bufB + big;          // agg / z      [G,L,128]
  float* deg   = bufC + big;
  float* dinv  = deg + LL;
  float* selfc = dinv + LL;
  float* norm  = selfc + LL;
  float* stats = norm + EE;                    // 4*CC: sum, sumsq, scale, shift
  float* mbuf  = stats + 4 * CC;               // [B, L, C] set-means (4 MiB)

  // x [G, C0, L] -> bufA [G, L, C0]
  k_transpose<<<dim3(LL / 32, C0IN / 32, GNUM), dim3(32, 8), 0, stream>>>(x, bufA, C0IN, LL);

  // degree / normalization coefficients
  k_zero<<<(LL + 255) / 256, 256, 0, stream>>>(deg, LL);
  k_deg<<<EE / 256, 256, 0, stream>>>(dstI, deg);
  k_dinv<<<LL / 256, 256, 0, stream>>>(deg, dinv, selfc);
  k_edgenorm<<<EE / 256, 256, 0, stream>>>(srcI, dstI, dinv, norm);

  const int gemmBlocks = GNUM * LL / 32;        // 4096 (32 M-rows per block)
  const int vecBlocks  = (int)(big / 4 / 256);  // 16384 (float4 granularity)

  for (int layer = 0; layer < 3; ++layer) {
    if (layer == 0)
      k_gemm_wmma<C0IN><<<gemmBlocks, 256, 0, stream>>>(bufA, Wl[0], selfc,
                                                        bl[0], bufB, bufC);
    else
      k_gemm_wmma<CC><<<gemmBlocks, 256, 0, stream>>>(bufA, Wl[layer], selfc,
                                                      bl[layer], bufB, bufC);

    k_scatter<<<GNUM * EE / 8, 256, 0, stream>>>((const float4*)bufB, srcI, dstI, norm, bufC);

    k_zero<<<1, 256, 0, stream>>>(stats, 2 * CC);
    k_center_stats<<<dim3(LL, BNUM), 128, 0, stream>>>(bufC, mbuf, stats);
    k_stats_final<<<1, 128, 0, stream>>>(stats, gml[layer], bel[layer]);
    k_apply<<<vecBlocks, 256, 0, stream>>>((const float4*)bufC, (const float4*)mbuf,
                                           stats, (float4*)bufA);
  }

  // bufA [G, L, 128] -> d_out [G, 128, L]
  k_transpose<<<dim3(CC / 32, LL / 32, GNUM), dim3(32, 8), 0, stream>>>(bufA, (float*)d_out, LL, CC);
}